// MXQuantizerSTE_35845797053136
// MI455X (gfx1250) — compile-verified
//
#include <hip/hip_runtime.h>

// ext-vector types (avoid HIP float4 struct so nontemporal builtins work)
typedef __attribute__((ext_vector_type(4))) float v4f;
typedef __attribute__((ext_vector_type(2))) float v2f;

// ---- FP8 E4M3 hardware conversion (RNE, subnormal-preserving) -------------
// Pack 4 floats into 4 E4M3 bytes of ONE register: lo pair -> word0, hi pair -> word1.
// Chaining the second convert through the first's result avoids per-iteration
// zero-materialization of the unwritten 16-bit halves (v_mov_b16 noise).
__device__ __forceinline__ unsigned pack4_fp8(float a, float b, float c, float d) {
#if __has_builtin(__builtin_amdgcn_cvt_pk_fp8_f32)
  int p = __builtin_amdgcn_cvt_pk_fp8_f32(a, b, 0, false);   // writes [15:0]
  p = __builtin_amdgcn_cvt_pk_fp8_f32(c, d, p, true);        // writes [31:16]
  return (unsigned)p;
#else
  unsigned r;
  asm volatile("v_cvt_pk_fp8_f32 %0, %1, %2" : "=v"(r) : "v"(a), "v"(b));
  asm volatile("v_cvt_pk_fp8_f32 %0, %1, %2 op_sel:[0,0,1]"
               : "+v"(r) : "v"(c), "v"(d));
  return r;
#endif
}

__device__ __forceinline__ v2f unpack2_fp8_lo(unsigned p) {
#if __has_builtin(__builtin_amdgcn_cvt_pk_f32_fp8)
  return __builtin_amdgcn_cvt_pk_f32_fp8((int)p, false);     // bytes 0,1
#else
  v2f r;
  asm volatile("v_cvt_pk_f32_fp8 %0, %1" : "=v"(r) : "v"(p));
  return r;
#endif
}

__device__ __forceinline__ v2f unpack2_fp8_hi(unsigned p) {
#if __has_builtin(__builtin_amdgcn_cvt_pk_f32_fp8)
  return __builtin_amdgcn_cvt_pk_f32_fp8((int)p, true);      // bytes 2,3
#else
  v2f r;
  asm volatile("v_cvt_pk_f32_fp8 %0, %1 op_sel:[1,0]" : "=v"(r) : "v"(p));
  return r;
#endif
}

// max over groups of 8 consecutive lanes (one MX block = 8 lanes x float4)
// ds_swizzle group-of-32 encoding: [14:10]=xor_mask, [9:5]=or_mask, [4:0]=and_mask
// SWAPX1 = 0x041f, SWAPX2 = 0x081f, SWAPX4 = 0x101f  (ISA ch.11 / ds_swizzle_b32)
__device__ __forceinline__ float group8_max(float a) {
  a = fmaxf(a, __int_as_float(__builtin_amdgcn_ds_swizzle(__float_as_int(a), 0x041f)));
  a = fmaxf(a, __int_as_float(__builtin_amdgcn_ds_swizzle(__float_as_int(a), 0x081f)));
  a = fmaxf(a, __int_as_float(__builtin_amdgcn_ds_swizzle(__float_as_int(a), 0x101f)));
  return a;
}

__global__ __launch_bounds__(256, 1) void mx_qdq_kernel(
    const float* __restrict__ x, float* __restrict__ y,
    unsigned long long n4 /* count of float4, multiple of 8 */) {
  const v4f* __restrict__ xv = (const v4f*)x;
  v4f* __restrict__ yv = (v4f*)y;

  unsigned long long tid = (unsigned long long)blockIdx.x * blockDim.x + threadIdx.x;
  unsigned long long stride = (unsigned long long)gridDim.x * blockDim.x;

  for (unsigned long long i = tid; i < n4; i += stride) {
    // streaming 128-bit load, non-temporal (read-once data)
    v4f v = __builtin_nontemporal_load(xv + i);

    // per-lane amax over 4 elements, then over the 8-lane block group
    float a = fmaxf(fmaxf(__builtin_fabsf(v.x), __builtin_fabsf(v.y)),
                    fmaxf(__builtin_fabsf(v.z), __builtin_fabsf(v.w)));
    a = group8_max(a);

    // shared_exp = clip(floor(log2(amax)) - 8, -127, 127); amax==0 -> scale=1
    // frexp: a = m * 2^fe, m in [0.5,1)  =>  floor(log2(a)) = fe - 1
    // branchless: v_frexp_exp_i32_f32 + integer clamp + cndmask on zero block
    int fe = __builtin_amdgcn_frexp_expf(a);   // denormal-safe exponent
    int se = fe - 9;                           // (fe - 1) - ELEM_EMAX(8)
    se = se < -127 ? -127 : (se > 127 ? 127 : se);
    se = (a > 0.0f) ? se : 0;

    float scale = __builtin_amdgcn_ldexpf(1.0f, se);    // exact 2^se (incl. 2^-127)
    float inv   = __builtin_amdgcn_ldexpf(1.0f, -se);   // exact 2^-se, no division

    // saturate to E4M3 max (reference clamps magnitude at 448 before rounding)
    float tx = fminf(fmaxf(v.x * inv, -448.0f), 448.0f);
    float ty = fminf(fmaxf(v.y * inv, -448.0f), 448.0f);
    float tz = fminf(fmaxf(v.z * inv, -448.0f), 448.0f);
    float tw = fminf(fmaxf(v.w * inv, -448.0f), 448.0f);

    // hardware E4M3 quantize (RNE, subnormals preserved) and dequantize,
    // all 4 bytes packed in one VGPR
    unsigned p = pack4_fp8(tx, ty, tz, tw);
    v2f d01 = unpack2_fp8_lo(p);
    v2f d23 = unpack2_fp8_hi(p);

    v4f o;
    o.x = d01.x * scale;
    o.y = d01.y * scale;
    o.z = d23.x * scale;
    o.w = d23.y * scale;

    // streaming 128-bit store, non-temporal (never re-read)
    __builtin_nontemporal_store(o, yv + i);
  }
}

extern "C" void kernel_launch(void* const* d_in, const int* in_sizes, int n_in,
                              void* d_out, int out_size, void* d_ws, size_t ws_size,
                              hipStream_t stream) {
  const float* x = (const float*)d_in[0];
  float* y = (float*)d_out;

  long long n = (long long)in_sizes[0];      // 4096*8192 = 33,554,432 (mult. of 128)
  unsigned long long n4 = (unsigned long long)(n / 4);

  const int block = 256;                     // 8 wave32s per workgroup
  long long need = (long long)((n4 + block - 1) / block);
  int grid = (int)(need < 16384 ? need : 16384);
  if (grid < 1) grid = 1;

  mx_qdq_kernel<<<grid, block, 0, stream>>>(x, y, n4);
}